// GCNConv_19361712571372
// MI455X (gfx1250) — compile-verified
//
#include <hip/hip_runtime.h>

typedef float v2f __attribute__((ext_vector_type(2)));
typedef float v8f __attribute__((ext_vector_type(8)));

#define D_FEAT 128

// ---------------------------------------------------------------------------
// Kernel 1: zero the aggregation workspace
// ---------------------------------------------------------------------------
__global__ void gcn_zero_kernel(float* __restrict__ agg, int n) {
    int i = blockIdx.x * blockDim.x + threadIdx.x;
    if (i < n) agg[i] = 0.0f;
}

// ---------------------------------------------------------------------------
// Kernel 2: edge gather + scatter-add.
// One wave32 per edge. Lane L loads features[src[e]][4L..4L+3] as a float4
// (wave reads one contiguous 512B row -> global_load_b128, fully coalesced),
// then does 4 global_atomic_add_f32 into agg[dst[e]].
// ---------------------------------------------------------------------------
__global__ void gcn_scatter_kernel(const float* __restrict__ feat,
                                   const long long* __restrict__ src,
                                   const long long* __restrict__ dst,
                                   float* __restrict__ agg,
                                   int n_edges) {
    int tid  = blockIdx.x * blockDim.x + threadIdx.x;
    int e    = tid >> 5;
    int lane = tid & 31;
    if (e >= n_edges) return;

    long long s = src[e];
    long long d = dst[e];

    const float4 v = *(const float4*)(feat + (size_t)s * D_FEAT + lane * 4);
    float* o = agg + (size_t)d * D_FEAT + lane * 4;
    atomicAdd(o + 0, v.x);
    atomicAdd(o + 1, v.y);
    atomicAdd(o + 2, v.z);
    atomicAdd(o + 3, v.w);
}

// ---------------------------------------------------------------------------
// Kernel 3: out = agg @ W + b  via V_WMMA_F32_16X16X4_F32.
// Block = 256 threads = 8 waves. Block owns 64 output rows; wave w owns
// N-tile [16w, 16w+16). W (128x128, 64KB) and the 64x128 A strip (padded to
// stride 132 to avoid LDS bank conflicts on column reads) are staged in LDS.
//
// WMMA fp32 16x16x4 operand layout (ISA 7.12.2):
//   A (16x4):  lanes 0-15 row M=lane, {v0,v1}={K0,K1}; lanes 16-31 {K2,K3}
//   B (4x16):  lanes 0-15 col N=lane, {v0,v1}={K0,K1}; lanes 16-31 {K2,K3}
//   C/D (16x16): vgpr r, lanes 0-15 -> M=r,    N=lane
//                        lanes 16-31 -> M=r+8, N=lane-16
// ---------------------------------------------------------------------------
#define ROWS_PER_BLOCK 64
#define SA_STRIDE (D_FEAT + 4)

__global__ __launch_bounds__(256) void
gcn_gemm_kernel(const float* __restrict__ agg,
                const float* __restrict__ W,
                const float* __restrict__ bias,
                float* __restrict__ out,
                int n_rows) {
    __shared__ float sW[D_FEAT * D_FEAT];            // [k][n], 64 KB
    __shared__ float sA[ROWS_PER_BLOCK * SA_STRIDE]; // [m][k], padded, ~33 KB

    const int tid     = threadIdx.x;
    const int wave    = tid >> 5;       // 0..7 -> N tile
    const int lane    = tid & 31;
    const int rowBase = blockIdx.x * ROWS_PER_BLOCK;

    // Stage W: 16384 floats, 64 per thread, float4 vectorized, coalesced.
    for (int i = tid * 4; i < D_FEAT * D_FEAT; i += 256 * 4) {
        *(float4*)&sW[i] = *(const float4*)&W[i];
    }
    // Stage A strip: 64 rows x 128 floats, zero-padded past n_rows.
    for (int idx = tid; idx < ROWS_PER_BLOCK * (D_FEAT / 4); idx += 256) {
        int r = idx >> 5;            // 32 float4 per row
        int c = (idx & 31) * 4;
        float4 v = make_float4(0.0f, 0.0f, 0.0f, 0.0f);
        int gr = rowBase + r;
        if (gr < n_rows) v = *(const float4*)&agg[(size_t)gr * D_FEAT + c];
        *(float4*)&sA[r * SA_STRIDE + c] = v;
    }
    __syncthreads();

    const int nBase   = wave * 16;
    const int halfSel = lane >> 4;      // 0: K+{0,1}, 1: K+{2,3}
    const int l15     = lane & 15;
    const float bv    = bias[nBase + l15];

    for (int mSub = 0; mSub < ROWS_PER_BLOCK / 16; ++mSub) {
        const float* aRow = &sA[(mSub * 16 + l15) * SA_STRIDE];
        v8f acc = {};
#pragma unroll
        for (int k0 = 0; k0 < D_FEAT; k0 += 4) {
            const int ka = k0 + halfSel * 2;
            v2f a, b;
            a.x = aRow[ka];
            a.y = aRow[ka + 1];
            b.x = sW[ka * D_FEAT + nBase + l15];
            b.y = sW[(ka + 1) * D_FEAT + nBase + l15];
            acc = __builtin_amdgcn_wmma_f32_16x16x4_f32(
                /*neg_a=*/false, a, /*neg_b=*/false, b,
                /*c_mod=*/(short)0, acc, /*reuse_a=*/false, /*reuse_b=*/false);
        }
        const int mTileBase = rowBase + mSub * 16 + halfSel * 8;
#pragma unroll
        for (int r = 0; r < 8; ++r) {
            int m = mTileBase + r;
            if (m < n_rows) out[(size_t)m * D_FEAT + nBase + l15] = acc[r] + bv;
        }
    }
}

// ---------------------------------------------------------------------------
// kernel_launch
// Inputs: 0=features[N,128] f32, 1=src[E] i64, 2=dst[E] i64,
//         3=W[128,128] f32, 4=b[1,128] f32.  Output: [N,128] f32.
// ---------------------------------------------------------------------------
extern "C" void kernel_launch(void* const* d_in, const int* in_sizes, int n_in,
                              void* d_out, int out_size, void* d_ws, size_t ws_size,
                              hipStream_t stream) {
    const float*     feat = (const float*)d_in[0];
    const long long* src  = (const long long*)d_in[1];
    const long long* dst  = (const long long*)d_in[2];
    const float*     W    = (const float*)d_in[3];
    const float*     bias = (const float*)d_in[4];
    float*           out  = (float*)d_out;
    float*           agg  = (float*)d_ws;   // [n_nodes, 128] f32 scratch

    const int n_nodes = in_sizes[0] / D_FEAT;
    const int n_edges = in_sizes[1];
    const int agg_elems = n_nodes * D_FEAT;

    // 1) agg = 0
    gcn_zero_kernel<<<(agg_elems + 255) / 256, 256, 0, stream>>>(agg, agg_elems);

    // 2) scatter-add edge messages (wave per edge)
    {
        long long threads = (long long)n_edges * 32;
        int blocks = (int)((threads + 255) / 256);
        gcn_scatter_kernel<<<blocks, 256, 0, stream>>>(feat, src, dst, agg, n_edges);
    }

    // 3) out = agg @ W + b (WMMA fp32)
    {
        int blocks = (n_nodes + ROWS_PER_BLOCK - 1) / ROWS_PER_BLOCK;
        gcn_gemm_kernel<<<blocks, 256, 0, stream>>>(agg, W, bias, out, n_nodes);
    }
}